// StateVecSimTorch_33440615367196
// MI455X (gfx1250) — compile-verified
//
#include <hip/hip_runtime.h>
#include <hip/hip_bf16.h>

// 11-qubit statevector sim, batch=16, fused single kernel.
// One workgroup (256 threads = 8 wave32) per sample; state lives in LDS.
// Single-qubit layer applied as three 16x16 complex GEMMs via
// V_WMMA_F32_16X16X4_F32; CNOT ring collapsed to one GF(2)-linear
// permutation applied with LDS ping-pong.

#define NQ        11
#define DIM       2048          // 2^11
#define NL        4
#define NTHREADS  256
#define PER_TH    (DIM / NTHREADS)  // 8

typedef __attribute__((ext_vector_type(2))) float v2f;
typedef __attribute__((ext_vector_type(8))) float v8f;

struct SMem {
  float sRe[2][DIM];    // ping-pong state, real
  float sIm[2][DIM];    // ping-pong state, imag
  float gRe[3][256];    // three 16x16 group unitaries, real
  float gIm[3][256];    // three 16x16 group unitaries, imag
  float mRe[NQ][4];     // per-qubit 2x2 gates, real (row-major m00 m01 m10 m11)
  float mIm[NQ][4];
};

// State viewed as 16x128 for each group's contraction axis.
template <int G>
__device__ __forceinline__ int gaddr(int row, int col) {
  if (G == 0) {                 // rows = bits 10..7
    return (row << 7) | col;
  } else if (G == 1) {          // rows = bits 6..3
    return ((col >> 3) << 7) | (row << 3) | (col & 7);
  } else {                      // rows = bits 3..0
    return (col << 4) | row;
  }
}

// In-place S(16x128) <- G(16x16) * S(16x128), complex, via f32 WMMA.
// Wave w owns column tile w (cols 16w..16w+15); lanes own single columns,
// so the in-place update is race-free (per-wave DS ops are in order and
// all loads precede all stores).
template <int G>
__device__ __forceinline__ void apply_group(SMem* sm, float* sRe, float* sIm,
                                            int lane, int wave) {
  const int half = lane >> 4;        // 0 or 1
  const int n    = lane & 15;        // column within tile; also A row index m
  const int col  = (wave << 4) | n;

  v2f aRe[4], aIm[4], aImN[4], bRe[4], bIm[4];
#pragma unroll
  for (int j = 0; j < 4; ++j) {
    const int k0 = 4 * j + 2 * half; // per-lane K pair for this 16x16x4 step
    aRe[j].x = sm->gRe[G][n * 16 + k0];
    aRe[j].y = sm->gRe[G][n * 16 + k0 + 1];
    aIm[j].x = sm->gIm[G][n * 16 + k0];
    aIm[j].y = sm->gIm[G][n * 16 + k0 + 1];
    aImN[j]  = -aIm[j];              // f32 WMMA NEG only supports C; negate in VALU
    bRe[j].x = sRe[gaddr<G>(k0,     col)];
    bRe[j].y = sRe[gaddr<G>(k0 + 1, col)];
    bIm[j].x = sIm[gaddr<G>(k0,     col)];
    bIm[j].y = sIm[gaddr<G>(k0 + 1, col)];
  }

  v8f cRe = {0.f,0.f,0.f,0.f,0.f,0.f,0.f,0.f};
  v8f cIm = {0.f,0.f,0.f,0.f,0.f,0.f,0.f,0.f};
#pragma unroll
  for (int j = 0; j < 4; ++j) {
    // Re(out) += Re(G)Re(S) - Im(G)Im(S) ; Im(out) += Re(G)Im(S) + Im(G)Re(S)
    cRe = __builtin_amdgcn_wmma_f32_16x16x4_f32(false, aRe[j],  false, bRe[j],
                                                (short)0, cRe, false, false);
    cRe = __builtin_amdgcn_wmma_f32_16x16x4_f32(false, aImN[j], false, bIm[j],
                                                (short)0, cRe, false, false);
    cIm = __builtin_amdgcn_wmma_f32_16x16x4_f32(false, aRe[j],  false, bIm[j],
                                                (short)0, cIm, false, false);
    cIm = __builtin_amdgcn_wmma_f32_16x16x4_f32(false, aIm[j],  false, bRe[j],
                                                (short)0, cIm, false, false);
  }

#pragma unroll
  for (int v = 0; v < 8; ++v) {      // C/D layout: VGPR v -> rows v, v+8
    const int row = v + 8 * half;
    const int idx = gaddr<G>(row, col);
    sRe[idx] = cRe[v];
    sIm[idx] = cIm[v];
  }
}

__global__ __launch_bounds__(NTHREADS)
void qsim_kernel(const float* __restrict__ X, const float* __restrict__ W,
                 const float* __restrict__ Bi, float* __restrict__ out) {
  __shared__ SMem sm;
  const int tid  = threadIdx.x;
  const int b    = blockIdx.x;
  const int lane = tid & 31;
  const int wave = tid >> 5;

  // |0...0>
#pragma unroll
  for (int e = 0; e < PER_TH; ++e) {
    const int i = e * NTHREADS + tid;
    sm.sRe[0][i] = 0.f;
    sm.sIm[0][i] = 0.f;
  }
  if (tid == 0) sm.sRe[0][0] = 1.f;

  // Compose the 11-CNOT ring into one permutation: out[j] = in[src(j)].
  // state = U(10,0)...U(1,2)U(0,1) x  =>  src(j) = f(0,1)(f(1,2)(...f(10,0)(j)))
  int src[PER_TH];
#pragma unroll
  for (int e = 0; e < PER_TH; ++e) {
    int j = e * NTHREADS + tid;
#pragma unroll
    for (int k = NQ - 1; k >= 0; --k) {
      const int c = k, t = (k + 1) % NQ;           // CNOT (c -> t)
      j ^= ((j >> (NQ - 1 - c)) & 1) << (NQ - 1 - t);
    }
    src[e] = j;
  }

  int cb = 0;
  for (int l = 0; l < NL; ++l) {
    // ---- per-qubit Rot gates -------------------------------------------
    if (tid < NQ) {
      const int q = tid;
      float ang[3];
#pragma unroll
      for (int c = 0; c < 3; ++c) {
        const int ai = l * 33 + q * 3 + c;          // angle index, tiled X
        ang[c] = X[b * 8 + (ai & 7)] * W[(l * NQ + q) * 3 + c]
               + Bi[(l * NQ + q) * 3 + c];
      }
      const float ct  = cosf(0.5f * ang[1]);
      const float st  = sinf(0.5f * ang[1]);
      const float po  = 0.5f * (ang[0] + ang[2]);
      const float pm  = 0.5f * (ang[0] - ang[2]);
      const float cpo = cosf(po), spo = sinf(po);
      const float cpm = cosf(pm), spm = sinf(pm);
      sm.mRe[q][0] =  cpo * ct;  sm.mIm[q][0] = -spo * ct;   // m00
      sm.mRe[q][1] = -cpm * st;  sm.mIm[q][1] = -spm * st;   // m01
      sm.mRe[q][2] =  cpm * st;  sm.mIm[q][2] = -spm * st;   // m10
      sm.mRe[q][3] =  cpo * ct;  sm.mIm[q][3] =  spo * ct;   // m11
    }
    __syncthreads();

    // ---- build the three 16x16 group unitaries (one entry per thread) --
    {
      const int j = tid >> 4, k = tid & 15;
      // group 0: kron(M0..M3); qubit (0+bq) <-> bit (3-bq) of j,k
      {
        float re = 1.f, im = 0.f;
#pragma unroll
        for (int bq = 0; bq < 4; ++bq) {
          const int e = ((j >> (3 - bq)) & 1) * 2 + ((k >> (3 - bq)) & 1);
          const float mr = sm.mRe[bq][e], mi = sm.mIm[bq][e];
          const float nr = re * mr - im * mi, ni = re * mi + im * mr;
          re = nr; im = ni;
        }
        sm.gRe[0][tid] = re; sm.gIm[0][tid] = im;
      }
      // group 1: kron(M4..M7)
      {
        float re = 1.f, im = 0.f;
#pragma unroll
        for (int bq = 0; bq < 4; ++bq) {
          const int e = ((j >> (3 - bq)) & 1) * 2 + ((k >> (3 - bq)) & 1);
          const float mr = sm.mRe[4 + bq][e], mi = sm.mIm[4 + bq][e];
          const float nr = re * mr - im * mi, ni = re * mi + im * mr;
          re = nr; im = ni;
        }
        sm.gRe[1][tid] = re; sm.gIm[1][tid] = im;
      }
      // group 2: I2 (qubit 7, already done in group 1) (x) kron(M8..M10)
      {
        float re = 0.f, im = 0.f;
        if ((j >> 3) == (k >> 3)) {
          re = 1.f; im = 0.f;
#pragma unroll
          for (int bq = 0; bq < 3; ++bq) {
            const int e = ((j >> (2 - bq)) & 1) * 2 + ((k >> (2 - bq)) & 1);
            const float mr = sm.mRe[8 + bq][e], mi = sm.mIm[8 + bq][e];
            const float nr = re * mr - im * mi, ni = re * mi + im * mr;
            re = nr; im = ni;
          }
        }
        sm.gRe[2][tid] = re; sm.gIm[2][tid] = im;
      }
    }
    __syncthreads();

    // ---- apply the three group GEMMs in place (WMMA) -------------------
    float* curRe = sm.sRe[cb];
    float* curIm = sm.sIm[cb];
    apply_group<0>(&sm, curRe, curIm, lane, wave);
    __syncthreads();
    apply_group<1>(&sm, curRe, curIm, lane, wave);
    __syncthreads();
    apply_group<2>(&sm, curRe, curIm, lane, wave);
    __syncthreads();

    // ---- composed CNOT-ring permutation, ping-pong ---------------------
    float* nRe = sm.sRe[cb ^ 1];
    float* nIm = sm.sIm[cb ^ 1];
#pragma unroll
    for (int e = 0; e < PER_TH; ++e) {
      const int j = e * NTHREADS + tid;
      nRe[j] = curRe[src[e]];
      nIm[j] = curIm[src[e]];
    }
    __syncthreads();
    cb ^= 1;
  }

  // ---- <Z_q> for q = 0..3 ----------------------------------------------
  const float* fRe = sm.sRe[cb];
  const float* fIm = sm.sIm[cb];
  float z[4] = {0.f, 0.f, 0.f, 0.f};
#pragma unroll
  for (int e = 0; e < PER_TH; ++e) {
    const int i = e * NTHREADS + tid;
    const float p = fRe[i] * fRe[i] + fIm[i] * fIm[i];
#pragma unroll
    for (int q = 0; q < 4; ++q)
      z[q] += ((i >> (10 - q)) & 1) ? -p : p;
  }
  float* red = sm.sRe[cb ^ 1];   // reuse dead buffer (1024 floats needed)
#pragma unroll
  for (int q = 0; q < 4; ++q) red[q * NTHREADS + tid] = z[q];
  __syncthreads();
  for (int s = NTHREADS / 2; s > 0; s >>= 1) {
    if (tid < s) {
#pragma unroll
      for (int q = 0; q < 4; ++q)
        red[q * NTHREADS + tid] += red[q * NTHREADS + tid + s];
    }
    __syncthreads();
  }
  if (tid < 4) out[b * 4 + tid] = red[tid * NTHREADS];
}

extern "C" void kernel_launch(void* const* d_in, const int* in_sizes, int n_in,
                              void* d_out, int out_size, void* d_ws, size_t ws_size,
                              hipStream_t stream) {
  (void)n_in; (void)out_size; (void)d_ws; (void)ws_size;
  const float* X  = (const float*)d_in[0];   // (16, 8)
  const float* W  = (const float*)d_in[1];   // (4, 11, 3)
  const float* Bi = (const float*)d_in[2];   // (4, 11, 3)
  float* out = (float*)d_out;                // (16, 4)
  const int batch = in_sizes[0] / 8;         // 16
  qsim_kernel<<<batch, NTHREADS, 0, stream>>>(X, W, Bi, out);
}